// prompt_gcn_21534966022320
// MI455X (gfx1250) — compile-verified
//
#include <hip/hip_runtime.h>

typedef float v2f __attribute__((ext_vector_type(2)));
typedef float v8f __attribute__((ext_vector_type(8)));

#define FDIM 128

// ---- degree histogram: one thread per edge, two f32 atomics --------------
__global__ __launch_bounds__(256) void k_degrees(const int* __restrict__ src,
                                                 const int* __restrict__ dst,
                                                 float* __restrict__ dsrc,
                                                 float* __restrict__ ddst, int E) {
  int e = blockIdx.x * blockDim.x + threadIdx.x;
  if (e < E) {
    atomicAdd(&dsrc[src[e]], 1.0f);
    atomicAdd(&ddst[dst[e]], 1.0f);
  }
}

// ---- w = rsqrt(max(w,1)) over both degree arrays -------------------------
__global__ __launch_bounds__(256) void k_rsqrt(float* __restrict__ w, int n) {
  int i = blockIdx.x * blockDim.x + threadIdx.x;
  if (i < n) w[i] = rsqrtf(fmaxf(w[i], 1.0f));
}

// ---- COO SpMM scatter: out[dst] += h[src] * scale[src] -------------------
// one wave32 per edge, float4 per lane (32*16B = 512B = one 128-f32 row)
__global__ __launch_bounds__(256) void k_spmm(const float* __restrict__ h,
                                              const float* __restrict__ scale,
                                              const int* __restrict__ src,
                                              const int* __restrict__ dst,
                                              float* __restrict__ out, int E) {
  int gid  = blockIdx.x * blockDim.x + threadIdx.x;
  int e    = gid >> 5;
  int lane = gid & 31;
  if (e >= E) return;
  int s = src[e];
  int d = dst[e];
  float sc = scale ? scale[s] : 1.0f;
  const float4 v = ((const float4*)(h + (size_t)s * FDIM))[lane];
  float* o = out + (size_t)d * FDIM + lane * 4;
  atomicAdd(o + 0, v.x * sc);
  atomicAdd(o + 1, v.y * sc);
  atomicAdd(o + 2, v.z * sc);
  atomicAdd(o + 3, v.w * sc);
}

// ---- h[i] = h[i] * rw[row] + bias[col]  (layer-0 tail) -------------------
__global__ __launch_bounds__(256) void k_scale_bias(float* __restrict__ h,
                                                    const float* __restrict__ rw,
                                                    const float* __restrict__ bias,
                                                    int total) {
  int i = blockIdx.x * blockDim.x + threadIdx.x;
  if (i < total) h[i] = h[i] * rw[i >> 7] + bias[i & (FDIM - 1)];
}

// ---- layer-1 tail: Hout = (A * rowscale[row]) @ W + bias -----------------
// f32 WMMA 16x16x4. Block = 256 threads = 8 waves; wave w owns col tile
// w*16; block owns 16 rows. K=128 -> 32 chained WMMAs. N is a multiple of 16
// for this problem (100000 = 6250*16).
__global__ __launch_bounds__(256) void k_gemm(const float* __restrict__ A,
                                              const float* __restrict__ rowscale,
                                              const float* __restrict__ W,
                                              const float* __restrict__ bias,
                                              float* __restrict__ Hout, int N) {
  int wid  = threadIdx.x >> 5;   // 0..7 -> column tile
  int lane = threadIdx.x & 31;
  int r0   = blockIdx.x * 16;
  int c0   = wid * 16;
  if (r0 >= N) return;

  int m    = lane & 15;          // A row / B,C column within tile
  int koff = (lane >> 4) * 2;    // K sub-offset held by this lane half
  int row  = r0 + m;
  float s  = rowscale[row];

  const float* pa = A + (size_t)row * FDIM + koff;        // A[m][koff + k0 + {0,1}]
  const float* pb = W + (size_t)koff * FDIM + c0 + m;     // W[koff + k0 + {0,1}][n]

  // C/D layout: VGPR j, lanes 0-15 -> row j, lanes 16-31 -> row j+8,
  // col = lane&15 everywhere -> bias depends only on n: broadcast-init C.
  float bn = bias[c0 + m];
  v8f acc;
#pragma unroll
  for (int j = 0; j < 8; ++j) acc[j] = bn;

#pragma unroll
  for (int k0 = 0; k0 < FDIM; k0 += 4) {
    v2f a, b;
    a[0] = pa[k0] * s;
    a[1] = pa[k0 + 1] * s;
    b[0] = pb[(size_t)k0 * FDIM];
    b[1] = pb[(size_t)(k0 + 1) * FDIM];
    acc = __builtin_amdgcn_wmma_f32_16x16x4_f32(
        /*neg_a=*/false, a, /*neg_b=*/false, b,
        /*c_mod=*/(short)0, acc, /*reuse_a=*/false, /*reuse_b=*/false);
  }

  int orow = r0 + (lane >> 4) * 8;             // rows j (lanes<16) / j+8 (lanes>=16)
  float* po = Hout + (size_t)orow * FDIM + c0 + m;
#pragma unroll
  for (int j = 0; j < 8; ++j) po[(size_t)j * FDIM] = acc[j];
}

extern "C" void kernel_launch(void* const* d_in, const int* in_sizes, int n_in,
                              void* d_out, int out_size, void* d_ws, size_t ws_size,
                              hipStream_t stream) {
  const float* feat = (const float*)d_in[0];
  const int*   src  = (const int*)d_in[1];
  const int*   dst  = (const int*)d_in[2];
  const float* W1   = (const float*)d_in[3];
  const float* b0   = (const float*)d_in[4];
  const float* b1   = (const float*)d_in[5];
  const int N = in_sizes[0] / FDIM;
  const int E = in_sizes[1];

  // ws layout: [wsrc N][wdst N][t N*FDIM]  (52 MB)
  float* wsrc = (float*)d_ws;
  float* wdst = wsrc + N;
  float* t    = wdst + N;
  float* out  = (float*)d_out;

  hipMemsetAsync(d_ws, 0, ((size_t)2 * N + (size_t)N * FDIM) * sizeof(float), stream);
  hipMemsetAsync(d_out, 0, (size_t)N * FDIM * sizeof(float), stream);

  k_degrees<<<(E + 255) / 256, 256, 0, stream>>>(src, dst, wsrc, wdst, E);
  k_rsqrt<<<(2 * N + 255) / 256, 256, 0, stream>>>(wsrc, 2 * N);

  const int spmmBlocks = (E * 32 + 255) / 256;           // one wave per edge
  const int ewBlocks   = (int)(((size_t)N * FDIM + 255) / 256);

  // layer 0: t = SpMM(feat * wsrc);  t = t * wdst + b0   (== h0)
  k_spmm<<<spmmBlocks, 256, 0, stream>>>(feat, wsrc, src, dst, t, E);
  k_scale_bias<<<ewBlocks, 256, 0, stream>>>(t, wdst, b0, N * FDIM);

  // layer 1: d_out = SpMM(h0 * wsrc);  t = (d_out * wdst) @ W1 + b1  (== h1)
  k_spmm<<<spmmBlocks, 256, 0, stream>>>(t, wsrc, src, dst, out, E);
  k_gemm<<<(N + 15) / 16, 256, 0, stream>>>(out, wdst, W1, b1, t, N);

  // final: d_out = SpMM(h1), unnormalized
  hipMemsetAsync(d_out, 0, (size_t)N * FDIM * sizeof(float), stream);
  k_spmm<<<spmmBlocks, 256, 0, stream>>>(t, nullptr, src, dst, out, E);
}